// LearnableFlatPool_40106404610417
// MI455X (gfx1250) — compile-verified
//
#include <hip/hip_runtime.h>
#include <stdint.h>

// LearnableFlatPool (stride-2, 3x3 max-plus pool with per-channel parabolic bias)
// B=16 C=128 H=W=224 -> Ho=Wo=111.  Memory-bound: ~512MB @ 23.3TB/s => ~22us floor.
// Path: global_load_async_to_lds_b128 (ASYNCcnt DMA) -> s_wait_asynccnt -> barrier
//       -> ds_load_b128 compute -> non-temporal b128 stores.

#define B_      16
#define C_      128
#define H_      224
#define W_      224
#define HO_     111
#define WO_     111
#define TOUT_   16                 // output rows per block
#define INROWS_ (2 * TOUT_ + 1)    // 33 input rows staged
#define LSTRIDE_ 228               // LDS row stride in floats (912B, 16B aligned, pads lane-27 overread)
#define CPR_    56                 // 16B chunks per 224-float input row

typedef __attribute__((ext_vector_type(4))) float f4;
typedef __attribute__((ext_vector_type(4))) int   i4;

#if defined(__gfx1250__) && __has_builtin(__builtin_amdgcn_global_load_async_to_lds_b128)
#define HAVE_ASYNC_B128 1
typedef __attribute__((address_space(1))) i4 gi4;   // global int4
typedef __attribute__((address_space(3))) i4 li4;   // LDS int4
#endif

__global__ __launch_bounds__(256)
void flatpool_kernel(const float* __restrict__ f, const float* __restrict__ t,
                     float* __restrict__ out)
{
    __shared__ float smem[INROWS_ * LSTRIDE_];

    const int plane = blockIdx.y;            // b*C + c, 0..2047
    const int c     = plane & (C_ - 1);
    const int ho0   = blockIdx.x * TOUT_;
    const int row0  = 2 * ho0;

    // h[c] collapses to 3 values: center 0, edge -(1/t)^16, corner -(2/t)^16 = -65536*(1/t)^16
    const float tv = t[c];
    const float r  = 1.0f / tv;
    const float r2 = r * r, r4 = r2 * r2, r8 = r4 * r4;
    const float r16 = r8 * r8;
    const float e1 = -r16;              // edge bias
    const float e2 = -65536.0f * r16;   // corner bias

    const float* fp = f + (size_t)plane * (size_t)(H_ * W_);

    // ---- Stage input rows row0 .. row0+32 into LDS (async DMA, 16B granules) ----
    for (int i = threadIdx.x; i < INROWS_ * CPR_; i += 256) {
        const int lr = i / CPR_;
        const int ch = i - lr * CPR_;
        const int gr = row0 + lr;
        if (gr < H_) {
            const float* gsrc = fp + (size_t)gr * W_ + ch * 4;
            float*       ldst = &smem[lr * LSTRIDE_ + ch * 4];
#if defined(HAVE_ASYNC_B128)
            __builtin_amdgcn_global_load_async_to_lds_b128(
                (gi4*)gsrc, (li4*)ldst, 0, 0);
#else
            *(f4*)ldst = *(const f4*)gsrc;   // fallback: global_load_b128 + ds_store_b128
#endif
        }
    }
#if defined(__HIP_DEVICE_COMPILE__)
#if __has_builtin(__builtin_amdgcn_s_wait_asynccnt)
    __builtin_amdgcn_s_wait_asynccnt(0);
#else
    asm volatile("s_wait_asynccnt 0" ::: "memory");
#endif
#endif
    __syncthreads();

    // ---- Compute: wave w -> output rows 2w, 2w+1 of this tile; lane -> 4 output cols ----
    const int wave = threadIdx.x >> 5;
    const int lane = threadIdx.x & 31;
    const int wo0  = lane * 4;               // 32 lanes * 4 = 128 >= 111
    float* po = out + (size_t)plane * (size_t)(HO_ * WO_);

    if (wo0 < WO_) {
        #pragma unroll
        for (int rr = 0; rr < 2; ++rr) {
            const int lro = 2 * wave + rr;
            const int ho  = ho0 + lro;
            if (ho < HO_) {
                const float* pa = &smem[(2 * lro + 0) * LSTRIDE_ + 2 * wo0];
                const float* pb = &smem[(2 * lro + 1) * LSTRIDE_ + 2 * wo0];
                const float* pc = &smem[(2 * lro + 2) * LSTRIDE_ + 2 * wo0];
                float a[12], bm[12], cv[12];
                *(f4*)&a[0]  = ((const f4*)pa)[0]; *(f4*)&a[4]  = ((const f4*)pa)[1]; *(f4*)&a[8]  = ((const f4*)pa)[2];
                *(f4*)&bm[0] = ((const f4*)pb)[0]; *(f4*)&bm[4] = ((const f4*)pb)[1]; *(f4*)&bm[8] = ((const f4*)pb)[2];
                *(f4*)&cv[0] = ((const f4*)pc)[0]; *(f4*)&cv[4] = ((const f4*)pc)[1]; *(f4*)&cv[8] = ((const f4*)pc)[2];

                float o[4];
                #pragma unroll
                for (int j = 0; j < 4; ++j) {
                    const int k = 2 * j;
                    // corners get e2, edges get e1, center gets 0
                    const float corners = fmaxf(fmaxf(a[k],     a[k + 2]),
                                                fmaxf(cv[k],    cv[k + 2]));
                    const float edges   = fmaxf(fmaxf(a[k + 1], cv[k + 1]),
                                                fmaxf(bm[k],    bm[k + 2]));
                    o[j] = fmaxf(fmaxf(corners + e2, edges + e1), bm[k + 1]);
                }

                float* prow = po + (size_t)ho * WO_ + wo0;
                if (wo0 + 3 < WO_) {
                    // lanes 0..26: full 16B non-temporal store (output never re-read)
                    f4 v; v.x = o[0]; v.y = o[1]; v.z = o[2]; v.w = o[3];
                    __builtin_nontemporal_store(v, (f4*)prow);
                } else {
                    #pragma unroll
                    for (int j = 0; j < 4; ++j)
                        if (wo0 + j < WO_) __builtin_nontemporal_store(o[j], prow + j);
                }
            }
        }
    }
}

extern "C" void kernel_launch(void* const* d_in, const int* in_sizes, int n_in,
                              void* d_out, int out_size, void* d_ws, size_t ws_size,
                              hipStream_t stream)
{
    const float* f = (const float*)d_in[0];   // (16,128,224,224) f32
    const float* t = (const float*)d_in[1];   // (128,) f32
    float* out = (float*)d_out;               // (16,128,111,111) f32

    dim3 grid((HO_ + TOUT_ - 1) / TOUT_, B_ * C_, 1);   // (7, 2048)
    dim3 block(256, 1, 1);                              // 8 waves (wave32)
    flatpool_kernel<<<grid, block, 0, stream>>>(f, t, out);
}